// Chain_45509473468743
// MI455X (gfx1250) — compile-verified
//
#include <hip/hip_runtime.h>
#include <stdint.h>

#ifndef USE_ASYNC_LDS
#define USE_ASYNC_LDS 1    // obs staging: global_load_async_to_lds_b128
#endif
#ifndef USE_ASYNC_STORE
#define USE_ASYNC_STORE 1  // result drain: global_store_async_from_lds_b64
#endif

namespace {
constexpr int kL = 4096;      // sequence length
constexpr int kChunk = 128;   // positions per lane (32 lanes * 128 = 4096)
constexpr int kRow = 130;     // words per chunk row (128 + 2 pad): b32 replay conflict-free
                              // (stride%64==2 -> banks 2*lane+p distinct) and 8B-aligned b64 drain
constexpr int kObsRow = 132;  // ints per staged obs chunk (128 + 4 pad -> conflict-free b128 reads)
}

// cross-lane read: value of v in lane srcLane (wave32; index wraps mod 32, callers guard)
__device__ __forceinline__ float permf(float v, int srcLane) {
  return __int_as_float(__builtin_amdgcn_ds_bpermute(srcLane * 4, __float_as_int(v)));
}

__global__ __launch_bounds__(32) void chain_maxplus_kernel(
    const float* __restrict__ jp, const float* __restrict__ bp,
    const int* __restrict__ obs, float* __restrict__ out) {
  // Two per-state row buffers, 32 chunks x 130 words each = 33280 B total.
  // The first 16896 B are also reused for the coalesced obs staging.
  __shared__ float lds[2 * 32 * kRow];
  float* ldsA = lds;                // state s=0: fwd msg f0, later out row 0
  float* ldsB = lds + 32 * kRow;    // state s=1: fwd msg f1, later out row 1

  const int lane = threadIdx.x;  // blockDim.x == 32 (one wave per sequence)
  const int seq  = blockIdx.x;

  // psi = j * [[.25,-.25],[-.25,.25]]; phi_i = (-c_i, +c_i) with c_i = 0.5*b[obs_i]
  const float jv = jp[0];
  const float a  = 0.25f * jv;
  const float cA = 0.5f * bp[0];
  const float cB = 0.5f * bp[1];
  const float dA = a - cA, dB = a - cB;  // step uses d = a - c
  const float eA = a + cA, eB = a + cB;  // and     e = a + c

  const int* gsrc = obs + (size_t)seq * kL;
  int* iobs = (int*)lds;

  // ---- stage observations into LDS, coalesced (chunk c -> ints [c*132, c*132+128)) ----
#if USE_ASYNC_LDS
  #pragma unroll 8
  for (int it = 0; it < 32; ++it) {
    const int* gp = gsrc + it * kChunk + lane * 4;
    unsigned lo = (unsigned)(uintptr_t)(&iobs[it * kObsRow + lane * 4]);
    asm volatile("global_load_async_to_lds_b128 %0, %1, off"
                 :: "v"(lo), "v"(gp) : "memory");
  }
  asm volatile("s_wait_asynccnt 0" ::: "memory");
#else
  #pragma unroll 8
  for (int it = 0; it < 32; ++it) {
    int4 v = *(const int4*)(gsrc + it * kChunk + lane * 4);
    *(int4*)(&iobs[it * kObsRow + lane * 4]) = v;
  }
#endif
  __syncthreads();

  // ---- pack this lane's 128 observation bits into 4 VGPRs ----
  uint32_t wds[4] = {0u, 0u, 0u, 0u};
  #pragma unroll
  for (int k = 0; k < 32; ++k) {
    int4 v = *(const int4*)(&iobs[lane * kObsRow + k * 4]);
    uint32_t nib = (uint32_t)((v.x & 1) | ((v.y & 1) << 1) |
                              ((v.z & 1) << 2) | ((v.w & 1) << 3));
    wds[k >> 3] |= nib << ((k & 7) * 4);
  }
  __syncthreads();  // LDS now reused as the per-state message/output buffers

  // X = X (x) M(bit) in (max,+):  col0 uses +/-d, col1 uses -/+e
  auto stepmat = [&](float& x00, float& x01, float& x10, float& x11, int bit) {
    const float d = bit ? dB : dA;
    const float e = bit ? eB : eA;
    const float n00 = fmaxf(x00 + d, x01 - d);
    const float n01 = fmaxf(x00 - e, x01 + e);
    const float n10 = fmaxf(x10 + d, x11 - d);
    const float n11 = fmaxf(x10 - e, x11 + e);
    x00 = n00; x01 = n01; x10 = n10; x11 = n11;
  };

  // ======== forward chunk composite: C = M_{i0} (x) ... (x) M_{i0+127} ========
  float C00, C01, C10, C11;
  {
    const int bit = (int)(wds[0] & 1u);
    const float d = bit ? dB : dA, e = bit ? eB : eA;
    C00 = d; C01 = -e; C10 = -d; C11 = e;  // = M at position 0
  }
  #pragma unroll
  for (int w = 0; w < 4; ++w) {
    const uint32_t ww = wds[w];
    #pragma unroll 8
    for (int t = (w == 0) ? 1 : 0; t < 32; ++t)
      stepmat(C00, C01, C10, C11, (int)((ww >> t) & 1u));
  }

  // ======== wave inclusive scan (lane 0 -> 31) of composites ========
  #pragma unroll
  for (int off = 1; off < 32; off <<= 1) {
    const float A00 = permf(C00, lane - off);
    const float A01 = permf(C01, lane - off);
    const float A10 = permf(C10, lane - off);
    const float A11 = permf(C11, lane - off);
    if (lane >= off) {  // C = A (x) C  (A is earlier in the sequence)
      const float n00 = fmaxf(A00 + C00, A01 + C10);
      const float n01 = fmaxf(A00 + C01, A01 + C11);
      const float n10 = fmaxf(A10 + C00, A11 + C10);
      const float n11 = fmaxf(A10 + C01, A11 + C11);
      C00 = n00; C01 = n01; C10 = n10; C11 = n11;
    }
  }
  // exclusive prefix -> forward message entering this chunk (zero-vector applied)
  float f0, f1;
  {
    const float P00 = permf(C00, lane - 1);
    const float P01 = permf(C01, lane - 1);
    const float P10 = permf(C10, lane - 1);
    const float P11 = permf(C11, lane - 1);
    f0 = (lane == 0) ? 0.0f : fmaxf(P00, P10);
    f1 = (lane == 0) ? 0.0f : fmaxf(P01, P11);
  }

  // ======== forward replay: park fwd messages per state (conflict-free b32) ========
  #pragma unroll
  for (int w = 0; w < 4; ++w) {
    const uint32_t ww = wds[w];
    #pragma unroll 8
    for (int t = 0; t < 32; ++t) {
      const int p = w * 32 + t;
      ldsA[lane * kRow + p] = f0;
      ldsB[lane * kRow + p] = f1;
      const int bit = (int)((ww >> t) & 1u);
      const float d = bit ? dB : dA, e = bit ? eB : eA;
      const float n0 = fmaxf(f0 + d, f1 - d);
      const float n1 = fmaxf(f0 - e, f1 + e);
      f0 = n0; f1 = n1;
    }
  }

  // ======== backward chunk composite: R = M_{i0+127} (x) ... (x) M_{i0} ========
  float R00, R01, R10, R11;
  {
    const int bit = (int)((wds[3] >> 31) & 1u);
    const float d = bit ? dB : dA, e = bit ? eB : eA;
    R00 = d; R01 = -e; R10 = -d; R11 = e;  // = M at position 127
  }
  #pragma unroll
  for (int w = 3; w >= 0; --w) {
    const uint32_t ww = wds[w];
    #pragma unroll 8
    for (int t = (w == 3) ? 30 : 31; t >= 0; --t)
      stepmat(R00, R01, R10, R11, (int)((ww >> t) & 1u));
  }

  // ======== wave suffix scan (lane 31 -> 0) ========
  #pragma unroll
  for (int off = 1; off < 32; off <<= 1) {
    const float A00 = permf(R00, lane + off);
    const float A01 = permf(R01, lane + off);
    const float A10 = permf(R10, lane + off);
    const float A11 = permf(R11, lane + off);
    if (lane < 32 - off) {  // R = A (x) R  (A covers higher lanes = earlier backward factors)
      const float n00 = fmaxf(A00 + R00, A01 + R10);
      const float n01 = fmaxf(A00 + R01, A01 + R11);
      const float n10 = fmaxf(A10 + R00, A11 + R10);
      const float n11 = fmaxf(A10 + R01, A11 + R11);
      R00 = n00; R01 = n01; R10 = n10; R11 = n11;
    }
  }
  // exclusive suffix -> backward message at this chunk's rightmost position
  float g0, g1;
  {
    const float Q00 = permf(R00, lane + 1);
    const float Q01 = permf(R01, lane + 1);
    const float Q10 = permf(R10, lane + 1);
    const float Q11 = permf(R11, lane + 1);
    g0 = (lane == 31) ? 0.0f : fmaxf(Q00, Q10);
    g1 = (lane == 31) ? 0.0f : fmaxf(Q01, Q11);
  }

  // ======== backward replay fused with output combine: out = phi + fwd + bwd ========
  #pragma unroll
  for (int w = 3; w >= 0; --w) {
    const uint32_t ww = wds[w];
    #pragma unroll 8
    for (int t = 31; t >= 0; --t) {
      const int p = w * 32 + t;
      const float fv0 = ldsA[lane * kRow + p];
      const float fv1 = ldsB[lane * kRow + p];
      const int bit = (int)((ww >> t) & 1u);
      const float c = bit ? cB : cA;  // phi = (-c, +c)
      ldsA[lane * kRow + p] = fv0 + g0 - c;
      ldsB[lane * kRow + p] = fv1 + g1 + c;
      const float d = bit ? dB : dA, e = bit ? eB : eA;
      const float n0 = fmaxf(g0 + d, g1 - d);
      const float n1 = fmaxf(g0 - e, g1 + e);
      g0 = n0; g1 = n1;
    }
  }

  // ======== drain both output rows [2, L]: async LDS -> global (coalesced b64) ========
#if USE_ASYNC_STORE
  asm volatile("s_wait_dscnt 0" ::: "memory");  // replay LDS writes visible to async unit
#else
  __syncthreads();
#endif
  float* orow0 = out + (size_t)seq * (2 * kL);
  #pragma unroll 4
  for (int j2 = 0; j2 < 64; ++j2) {
    // output floats idx = j2*64 + 2*lane (+1); chunk = j2/2, p = (j2&1)*64 + 2*lane
    const int wbase = (j2 >> 1) * kRow + (j2 & 1) * 64 + lane * 2;
    float* ga = orow0 + j2 * 64 + lane * 2;  // row 0 (state 0)
    float* gb = ga + kL;                     // row 1 (state 1)
#if USE_ASYNC_STORE
    unsigned la = (unsigned)(uintptr_t)(&ldsA[wbase]);
    unsigned lb = (unsigned)(uintptr_t)(&ldsB[wbase]);
    asm volatile("global_store_async_from_lds_b64 %0, %1, off"
                 :: "v"(ga), "v"(la) : "memory");
    asm volatile("global_store_async_from_lds_b64 %0, %1, off"
                 :: "v"(gb), "v"(lb) : "memory");
#else
    const float2 va = *(const float2*)(&ldsA[wbase]);
    const float2 vb = *(const float2*)(&ldsB[wbase]);
    *(float2*)ga = va;
    *(float2*)gb = vb;
#endif
  }
  // s_endpgm performs an implicit wait-idle, draining the async stores.
}

extern "C" void kernel_launch(void* const* d_in, const int* in_sizes, int n_in,
                              void* d_out, int out_size, void* d_ws, size_t ws_size,
                              hipStream_t stream) {
  const float* jp  = (const float*)d_in[0];
  const float* bp  = (const float*)d_in[1];
  const int*   obs = (const int*)d_in[2];
  float*       out = (float*)d_out;
  const int B = in_sizes[2] / kL;  // 2048 sequences -> 2048 single-wave workgroups
  chain_maxplus_kernel<<<B, 32, 0, stream>>>(jp, bp, obs, out);
}